// PoseEstimation_52029233824208
// MI455X (gfx1250) — compile-verified
//
#include <hip/hip_runtime.h>
#include <hip/hip_bf16.h>

// ---------------------------------------------------------------------------
// PoseEstimation restructured for MI455X (gfx1250, wave32):
//  - MLP is affine (no activations)  ->  fold w1@w2@w3@w4 once (WMMA f32 GEMM,
//    B staged zero-padded in LDS so the K-loop has no EXEC divergence)
//  - conv(5,stride5,VALID) on binary voxel grid -> per-voxel 7-vector table E
//  - per iteration: LDS-staged bitmask voxelize + E-table reduce + quat update
//  - loss = popcount(S ^ T) / (B*32^3)
// ---------------------------------------------------------------------------

#define BATCH 32
#define NPTS  131072
#define GRID_WORDS 1024          // 32^3 bits / 32
#define NVOX 27000               // 30*30*30 voxels covered by the conv
#define FIN_HALF 6912            // 32 ch * 6*6*6
#define KCHUNK 512               // B-panel rows staged in LDS per pass

typedef float v2f __attribute__((ext_vector_type(2)));
typedef float v8f __attribute__((ext_vector_type(8)));

// --------------------------- generic clear ---------------------------------
__global__ void k_clear_u32(unsigned* __restrict__ p, int n) {
    int i = blockIdx.x * blockDim.x + threadIdx.x;
    if (i < n) p[i] = 0u;
}

// ------------------- WMMA f32 GEMM: C[M x 8] = A[MxK] @ B[Kxn] -------------
// 256 threads = 8 waves = 8 x 16-row tiles per block. Single 16-col tile;
// B is staged in LDS zero-padded to 16 columns (cols >= ncolsB are 0), so the
// inner loop is branch-free and EXEC stays all-ones at every WMMA (ISA 7.12).
// Fragment layouts per CDNA5 ISA 7.12.2 (32-bit A 16x4 / 32-bit C 16x16).
__global__ void k_gemm_wmma(const float* __restrict__ A, int lda,
                            const float* __restrict__ B, int ldb, int ncolsB,
                            float* __restrict__ C, int ldc, int M, int K) {
    __shared__ float sB[KCHUNK * 16];            // 32 KB
    int t    = threadIdx.x;
    int wave = t >> 5;
    int lane = t & 31;
    int tile = blockIdx.x * (blockDim.x >> 5) + wave;
    int mt   = (M >> 4) - 1;
    if (tile > mt) tile = mt;                    // clamp (keeps barriers uniform)
    int m0   = tile * 16;
    int half = lane >> 4;                        // 0 | 1
    int l16  = lane & 15;
    int am   = m0 + l16;
    v8f acc = {};
    for (int kc = 0; kc < K; kc += KCHUNK) {
        int kl = K - kc;
        if (kl > KCHUNK) kl = KCHUNK;
        __syncthreads();                         // protect previous panel
        for (int idx = t; idx < kl * 16; idx += 256) {
            int row = idx >> 4, col = idx & 15;
            sB[idx] = (col < ncolsB) ? B[(size_t)(kc + row) * ldb + col] : 0.0f;
        }
        __syncthreads();
        for (int k0 = 0; k0 < kl; k0 += 4) {
            int ka = k0 + 2 * half;              // VGPR r -> K = 2*(lane/16)+r
            v2f a, bf;
            a.x  = A[(size_t)am * lda + (kc + ka) + 0];
            a.y  = A[(size_t)am * lda + (kc + ka) + 1];
            bf.x = sB[(ka + 0) * 16 + l16];      // branch-free padded B
            bf.y = sB[(ka + 1) * 16 + l16];
            acc = __builtin_amdgcn_wmma_f32_16x16x4_f32(
                false, a, false, bf, (short)0, acc, false, false);
        }
    }
    // C/D: VGPR r holds (M = m0 + r + 8*half, N = l16); keep first 8 columns.
    if (l16 < 8) {
#pragma unroll
        for (int r = 0; r < 8; ++r)
            C[(size_t)(m0 + r + 8 * half) * ldc + l16] = acc[r];
    }
}

// -------------------- folded bias: bc = ((b1@w2+b2)@w3+b3)@w4+b4 -----------
__global__ void k_bias_chain(const float* __restrict__ b1, const float* __restrict__ w2,
                             const float* __restrict__ b2, const float* __restrict__ w3,
                             const float* __restrict__ b3, const float* __restrict__ w4,
                             const float* __restrict__ b4, float* __restrict__ bc) {
    __shared__ float u[512];
    __shared__ float v[256];
    int t = threadIdx.x;                 // 512 threads
    if (t < 512) {
        float s = b2[t];
        for (int i = 0; i < 1024; ++i) s += b1[i] * w2[i * 512 + t];
        u[t] = s;
    }
    __syncthreads();
    if (t < 256) {
        float s = b3[t];
        for (int i = 0; i < 512; ++i) s += u[i] * w3[i * 256 + t];
        v[t] = s;
    }
    __syncthreads();
    if (t < 8) {
        float s = 0.0f;
        if (t < 7) {
            s = b4[t];
            for (int i = 0; i < 256; ++i) s += v[i] * w4[i * 7 + t];
        }
        bc[t] = s;
    }
}

// ------------- E table: per-voxel 8-vector (conv ⊗ collapsed MLP) ----------
__global__ void k_build_E(const float* __restrict__ conv_w, const float* __restrict__ Wc,
                          int wcRowOff, float* __restrict__ E) {
    int v = blockIdx.x * blockDim.x + threadIdx.x;
    if (v >= NVOX) return;
    int a = v / 900, rem = v - a * 900;
    int y = rem / 30, z = rem - y * 30;
    int od = a / 5, ka = a - od * 5;
    int oh = y / 5, kb = y - oh * 5;
    int ow = z / 5, kc = z - ow * 5;
    int pos  = (od * 6 + oh) * 6 + ow;         // feature position within channel
    int koff = ka * 25 + kb * 5 + kc;          // kernel offset
    float acc[8] = {0, 0, 0, 0, 0, 0, 0, 0};
    for (int c = 0; c < 32; ++c) {
        float w = conv_w[c * 125 + koff];
        const float* row = &Wc[(size_t)(wcRowOff + c * 216 + pos) * 8];
#pragma unroll
        for (int j = 0; j < 8; ++j) acc[j] += w * row[j];
    }
#pragma unroll
    for (int j = 0; j < 8; ++j) E[(size_t)v * 8 + j] = acc[j];
}

// ------ constant term: conv_b through both feature halves of Wc, + bc ------
__global__ void k_build_const(const float* __restrict__ conv_b, const float* __restrict__ Wc,
                              const float* __restrict__ bc, float* __restrict__ constj) {
    __shared__ float red[8 * 256];
    int t = threadIdx.x;                 // 256 threads
    float acc[8] = {0, 0, 0, 0, 0, 0, 0, 0};
    for (int r = t; r < FIN_HALF; r += 256) {
        float cb = conv_b[r / 216];
#pragma unroll
        for (int j = 0; j < 8; ++j)
            acc[j] += cb * (Wc[(size_t)r * 8 + j] + Wc[(size_t)(FIN_HALF + r) * 8 + j]);
    }
#pragma unroll
    for (int j = 0; j < 8; ++j) red[j * 256 + t] = acc[j];
    __syncthreads();
    for (int s = 128; s > 0; s >>= 1) {
        if (t < s)
#pragma unroll
            for (int j = 0; j < 8; ++j) red[j * 256 + t] += red[j * 256 + t + s];
        __syncthreads();
    }
    if (t < 8) constj[t] = red[t * 256] + bc[t];
}

// ----------------------------- voxelize ------------------------------------
__device__ __forceinline__ int vox_idx(float v) {
    int i = (int)floorf((v + 0.5f) * 32.0f);
    return i < 0 ? 0 : (i > 31 ? 31 : i);
}

// 1024 threads, 4 points/thread (3x float4 = 48B contiguous per thread),
// LDS-staged bitmask, one global atomicOr per word per block.
// Macc = identity reproduces voxelize(src) bit-exactly (x*1+y*0+z*0 == x).
__global__ void k_voxelize_rot(const float* __restrict__ pts, const float* __restrict__ Macc,
                               unsigned* __restrict__ bits) {
    __shared__ unsigned smask[GRID_WORDS];
    int b = blockIdx.y;
    int t = threadIdx.x;                 // 1024
    smask[t] = 0u;
    __syncthreads();
    const float* M = &Macc[b * 9];
    float m0 = M[0], m1 = M[1], m2 = M[2];
    float m3 = M[3], m4 = M[4], m5 = M[5];
    float m6 = M[6], m7 = M[7], m8 = M[8];
    size_t base = ((size_t)b * NPTS + (size_t)blockIdx.x * 4096 + (size_t)t * 4) * 3;
    const float4* p4 = (const float4*)(pts + base);
    float4 A = p4[0], Bv = p4[1], C = p4[2];
    float px[4] = {A.x, A.w, Bv.z, C.y};
    float py[4] = {A.y, Bv.x, Bv.w, C.z};
    float pz[4] = {A.z, Bv.y, C.x, C.w};
#pragma unroll
    for (int k = 0; k < 4; ++k) {
        float x = px[k], y = py[k], z = pz[k];
        float rx = m0 * x + m1 * y + m2 * z;
        float ry = m3 * x + m4 * y + m5 * z;
        float rz = m6 * x + m7 * y + m8 * z;
        int flat = (vox_idx(rx) << 10) + (vox_idx(ry) << 5) + vox_idx(rz);
        atomicOr(&smask[flat >> 5], 1u << (flat & 31));
    }
    __syncthreads();
    unsigned w = smask[t];
    if (w) atomicOr(&bits[(b << 10) + t], w);
}

// ------------- reduce E over occupied voxels (one block per batch) ---------
__global__ void k_reduce_E(const unsigned* __restrict__ bits, const float* __restrict__ E,
                           const float* __restrict__ add, int addPerBatch,
                           float* __restrict__ out) {
    int b = blockIdx.x;
    __shared__ unsigned sb[GRID_WORDS];
    __shared__ float red[8 * 256];
    int t = threadIdx.x;                 // 256 threads
    for (int i = t; i < GRID_WORDS; i += 256) sb[i] = bits[(b << 10) + i];
    __syncthreads();
    float acc[8] = {0, 0, 0, 0, 0, 0, 0, 0};
    for (int v = t; v < NVOX; v += 256) {
        int a = v / 900, rem = v - a * 900;
        int y = rem / 30, z = rem - y * 30;
        int flat = (a << 10) + (y << 5) + z;
        float occ = (float)((sb[flat >> 5] >> (flat & 31)) & 1u);
        const float* e = &E[(size_t)v * 8];
#pragma unroll
        for (int j = 0; j < 8; ++j) acc[j] += occ * e[j];   // ~98% occupancy: stay dense
    }
#pragma unroll
    for (int j = 0; j < 8; ++j) red[j * 256 + t] = acc[j];
    __syncthreads();
    for (int s = 128; s > 0; s >>= 1) {      // deterministic tree reduce
        if (t < s)
#pragma unroll
            for (int j = 0; j < 8; ++j) red[j * 256 + t] += red[j * 256 + t + s];
        __syncthreads();
    }
    if (t < 8) out[b * 8 + t] = red[t * 256] + (addPerBatch ? add[b * 8 + t] : add[t]);
}

// -------------------- quat -> R, Macc = R @ Macc ---------------------------
__global__ void k_init_macc(float* __restrict__ Macc) {
    int b = threadIdx.x;
    if (b >= BATCH) return;
#pragma unroll
    for (int i = 0; i < 9; ++i) Macc[b * 9 + i] = (i % 4 == 0) ? 1.0f : 0.0f;
}

__global__ void k_update_pose(const float* __restrict__ del, float* __restrict__ Macc) {
    int b = threadIdx.x;
    if (b >= BATCH) return;
    const float* d = &del[b * 8];
    float q0 = d[3], q1 = d[4], q2 = d[5], q3 = d[6];
    float R[9];
    R[0] = q0 * q0 + q1 * q1 - q2 * q2 - q3 * q3;
    R[1] = 2.0f * (q1 * q2 - q0 * q3);
    R[2] = 2.0f * (q1 * q3 + q0 * q2);
    R[3] = 2.0f * (q1 * q2 + q0 * q3);
    R[4] = q0 * q0 + q2 * q2 - q1 * q1 - q3 * q3;
    R[5] = 2.0f * (q2 * q3 - q0 * q1);
    R[6] = 2.0f * (q1 * q3 - q0 * q2);
    R[7] = 2.0f * (q2 * q3 + q0 * q1);
    R[8] = q0 * q0 + q3 * q3 - q1 * q1 - q2 * q2;
    float Mo[9], Mn[9];
#pragma unroll
    for (int i = 0; i < 9; ++i) Mo[i] = Macc[b * 9 + i];
#pragma unroll
    for (int i = 0; i < 3; ++i)
#pragma unroll
        for (int j = 0; j < 3; ++j)
            Mn[i * 3 + j] = R[i * 3 + 0] * Mo[0 * 3 + j] +
                            R[i * 3 + 1] * Mo[1 * 3 + j] +
                            R[i * 3 + 2] * Mo[2 * 3 + j];
#pragma unroll
    for (int i = 0; i < 9; ++i) Macc[b * 9 + i] = Mn[i];
}

// -------------------------- loss: popcount(S^T) ----------------------------
__global__ void k_xor_popcount(const unsigned* __restrict__ S, const unsigned* __restrict__ T,
                               unsigned* __restrict__ cnt) {
    __shared__ unsigned red[256];
    int i = blockIdx.x * 256 + threadIdx.x;
    red[threadIdx.x] = (unsigned)__popc(S[i] ^ T[i]);
    __syncthreads();
    for (int s = 128; s > 0; s >>= 1) {
        if (threadIdx.x < (unsigned)s) red[threadIdx.x] += red[threadIdx.x + s];
        __syncthreads();
    }
    if (threadIdx.x == 0) atomicAdd(cnt, red[0]);   // integer atomics: deterministic
}

__global__ void k_finalize(const unsigned* __restrict__ cnt, float* __restrict__ out) {
    if (threadIdx.x == 0 && blockIdx.x == 0)
        out[0] = (float)cnt[0] * (1.0f / (float)(BATCH * 32768));
}

// ---------------------------------------------------------------------------
extern "C" void kernel_launch(void* const* d_in, const int* in_sizes, int n_in,
                              void* d_out, int out_size, void* d_ws, size_t ws_size,
                              hipStream_t stream) {
    const float* src    = (const float*)d_in[0];
    const float* tmpl   = (const float*)d_in[1];
    const float* conv_w = (const float*)d_in[2];
    const float* conv_b = (const float*)d_in[3];
    const float* w1 = (const float*)d_in[4];
    const float* b1 = (const float*)d_in[5];
    const float* w2 = (const float*)d_in[6];
    const float* b2 = (const float*)d_in[7];
    const float* w3 = (const float*)d_in[8];
    const float* b3 = (const float*)d_in[9];
    const float* w4 = (const float*)d_in[10];
    const float* b4 = (const float*)d_in[11];
    const int maxItr = 8;   // setup_inputs(): python scalar 8 (device read not capturable)

    // workspace bump allocator (total ~2.5 MB)
    char* ws = (char*)d_ws;
    size_t off = 0;
    auto alloc = [&](size_t bytes) -> void* {
        void* p = ws + off;
        off = (off + bytes + 255) & ~(size_t)255;
        return p;
    };
    float*    t1     = (float*)alloc(512 * 8 * sizeof(float));
    float*    t2     = (float*)alloc(1024 * 8 * sizeof(float));
    float*    Wc     = (float*)alloc((size_t)13824 * 8 * sizeof(float));
    float*    bc     = (float*)alloc(8 * sizeof(float));
    float*    constj = (float*)alloc(8 * sizeof(float));
    float*    E_src  = (float*)alloc((size_t)NVOX * 8 * sizeof(float));
    float*    E_tmp  = (float*)alloc((size_t)NVOX * 8 * sizeof(float));
    unsigned* Tbits  = (unsigned*)alloc(BATCH * GRID_WORDS * sizeof(unsigned));
    unsigned* Sbits  = (unsigned*)alloc(BATCH * GRID_WORDS * sizeof(unsigned));
    float*    D0     = (float*)alloc(BATCH * 8 * sizeof(float));
    float*    del    = (float*)alloc(BATCH * 8 * sizeof(float));
    float*    Macc   = (float*)alloc(BATCH * 9 * sizeof(float));
    unsigned* cnt    = (unsigned*)alloc(sizeof(unsigned));
    (void)off; (void)ws_size; (void)in_sizes; (void)n_in; (void)out_size;

    // ---- one-time precompute: collapse MLP with WMMA f32 GEMMs ----
    // t1 = w3 @ w4           (512x256 @ 256x7  -> 512x8)
    k_gemm_wmma<<<512 / 16 / 8, 256, 0, stream>>>(w3, 256, w4, 7, 7, t1, 8, 512, 256);
    // t2 = w2 @ t1           (1024x512 @ 512x8 -> 1024x8)
    k_gemm_wmma<<<1024 / 16 / 8, 256, 0, stream>>>(w2, 512, t1, 8, 8, t2, 8, 1024, 512);
    // Wc = w1 @ t2           (13824x1024 @ 1024x8 -> 13824x8)
    k_gemm_wmma<<<13824 / 16 / 8, 256, 0, stream>>>(w1, 1024, t2, 8, 8, Wc, 8, 13824, 1024);
    k_bias_chain<<<1, 512, 0, stream>>>(b1, w2, b2, w3, b3, w4, b4, bc);

    // ---- per-voxel effect tables (conv folded with Wc) ----
    k_build_E<<<(NVOX + 255) / 256, 256, 0, stream>>>(conv_w, Wc, 0, E_src);
    k_build_E<<<(NVOX + 255) / 256, 256, 0, stream>>>(conv_w, Wc, FIN_HALF, E_tmp);
    k_build_const<<<1, 256, 0, stream>>>(conv_b, Wc, bc, constj);

    // ---- template grid + per-batch constant D0 (Macc == identity here) ----
    k_init_macc<<<1, 32, 0, stream>>>(Macc);
    k_clear_u32<<<(BATCH * GRID_WORDS + 255) / 256, 256, 0, stream>>>(Tbits, BATCH * GRID_WORDS);
    dim3 gpts(NPTS / 4096, BATCH);
    k_voxelize_rot<<<gpts, 1024, 0, stream>>>(tmpl, Macc, Tbits);
    k_reduce_E<<<BATCH, 256, 0, stream>>>(Tbits, E_tmp, constj, 0, D0);

    // ---- iterations ----
    for (int it = 0; it < maxItr; ++it) {
        k_clear_u32<<<(BATCH * GRID_WORDS + 255) / 256, 256, 0, stream>>>(Sbits, BATCH * GRID_WORDS);
        k_voxelize_rot<<<gpts, 1024, 0, stream>>>(src, Macc, Sbits);
        k_reduce_E<<<BATCH, 256, 0, stream>>>(Sbits, E_src, D0, 1, del);
        k_update_pose<<<1, 32, 0, stream>>>(del, Macc);
    }

    // ---- final loss ----
    k_clear_u32<<<(BATCH * GRID_WORDS + 255) / 256, 256, 0, stream>>>(Sbits, BATCH * GRID_WORDS);
    k_clear_u32<<<1, 32, 0, stream>>>(cnt, 1);
    k_voxelize_rot<<<gpts, 1024, 0, stream>>>(src, Macc, Sbits);
    k_xor_popcount<<<BATCH * GRID_WORDS / 256, 256, 0, stream>>>(Sbits, Tbits, cnt);
    k_finalize<<<1, 32, 0, stream>>>(cnt, (float*)d_out);
}